// CTCLossLayer_38508676776608
// MI455X (gfx1250) — compile-verified
//
#include <hip/hip_runtime.h>
#include <hip/hip_bf16.h>

// CTC forward loss, CDNA5/gfx1250.
// B=32, T=1000, V=1024 (blank=V-1), L=100, S=2L+1=201.
// One workgroup per batch element; thread s owns DP lane s.
// Probability rows (4KB each) stream through an 8-deep LDS ring via the CDNA5
// async global->LDS path (global_load_async_to_lds_b128, ASYNCcnt). Prefetch
// distance 7 hides HBM latency behind the serial log-semiring scan; the wait
// at each step is s_wait_asynccnt 6 (oldest-of-7 complete, in-order per wave).
// Alpha recurrence runs in log2 domain so v_exp_f32/v_log_f32 are used raw.

#define CTC_B 32
#define CTC_T 1000
#define CTC_V 1024
#define CTC_L 100
#define CTC_S 201     // 2*L+1
#define NBUF  8       // LDS row ring depth (power of two)

#define NEGV  (-1e9f)
#define EPSV  (1e-7f)
#define LN2F  (0.6931471805599453f)

// ---- CDNA5-specific primitives (inline asm; portable across toolchains) ----

// Workgroup-relative LDS byte offset: generic LDS addresses are
// {aperture[63:32], offset[31:0]}, so truncation yields the DS offset.
__device__ __forceinline__ unsigned ctc_lds_off(const void* p) {
    return (unsigned)(unsigned long long)p;
}

// GLOBAL_LOAD_ASYNC_TO_LDS_B128: VDST = per-lane LDS byte offset, VADDR = 64b
// global address (GV mode). Tracked with ASYNCcnt; completes in order.
__device__ __forceinline__ void ctc_async_ldg_b128(unsigned lds_byte_off,
                                                   const float* gaddr) {
    asm volatile("global_load_async_to_lds_b128 %0, %1, off"
                 :
                 : "v"(lds_byte_off), "v"(gaddr)
                 : "memory");
}

__device__ __forceinline__ void ctc_wait_async_le6() {
    asm volatile("s_wait_asynccnt 6" ::: "memory");
}
__device__ __forceinline__ void ctc_wait_async0() {
    asm volatile("s_wait_asynccnt 0" ::: "memory");
}

// Base-2 exp/log mapping straight to v_exp_f32 / v_log_f32.
// Fallbacks are host+device-safe libm names (only the builtin branch is taken
// in the device pass; the host pass never executes these).
__device__ __forceinline__ float ctc_exp2(float x) {
#if __has_builtin(__builtin_amdgcn_exp2f)
    return __builtin_amdgcn_exp2f(x);
#else
    return exp2f(x);
#endif
}
__device__ __forceinline__ float ctc_log2(float x) {
#if __has_builtin(__builtin_amdgcn_logf)
    return __builtin_amdgcn_logf(x);
#else
    return log2f(x);
#endif
}

// ---------------------------------------------------------------------------

__global__ __launch_bounds__(256) void ctc_alpha_kernel(
    const float* __restrict__ pred,       // [B,T,V] probabilities
    const int*   __restrict__ input_len,  // [B]
    const int*   __restrict__ labels,     // [B,L]
    const int*   __restrict__ label_len,  // [B]
    float*       __restrict__ loss_ws)    // [B] per-utterance loss out
{
    __shared__ float row[NBUF][CTC_V];    // 32 KB streaming ring
    __shared__ float alphaA[CTC_S];
    __shared__ float alphaB[CTC_S];

    const int b   = blockIdx.x;
    const int tid = threadIdx.x;
    const float* base = pred + (size_t)b * CTC_T * CTC_V;
    const float* gcol = base + tid * 4;   // this thread's 16B slice of a row

    // Prologue: issue async loads for rows 0..NBUF-2 (7 outstanding).
#pragma unroll
    for (int r = 0; r < NBUF - 1; ++r) {
        ctc_async_ldg_b128(ctc_lds_off(&row[r][tid * 4]),
                           gcol + (size_t)r * CTC_V);
    }

    const int il     = input_len[b];
    const int ll     = label_len[b];
    const int Svalid = 2 * ll + 1;        // lanes < Svalid are reachable

    // Per-lane extended symbol + skip flag (no cross-thread sharing needed).
    int myExt  = CTC_V - 1;               // blank on even lanes
    int skipOk = 0;
    if (tid < CTC_S && (tid & 1)) {
        myExt = labels[b * CTC_L + (tid >> 1)];
        if (tid >= 2)
            skipOk = (myExt != labels[b * CTC_L + (tid >> 1) - 1]) ? 1 : 0;
        // tid==1: reference allows skip but alpha[-2] lane is off-grid -> NEG anyway
    }

    ctc_wait_async_le6();                 // row 0 landed (in-order completion)
    __syncthreads();

    // Keep the pipeline full: issue row NBUF-1.
    ctc_async_ldg_b128(ctc_lds_off(&row[NBUF - 1][tid * 4]),
                       gcol + (size_t)(NBUF - 1) * CTC_V);

    float* cur = alphaA;
    float* nxt = alphaB;

    // t = 0 init (log2 domain).
    float myA = NEGV;
    if (tid < CTC_S) {
        if (tid == 0) {
            myA = ctc_log2(row[0][CTC_V - 1] + EPSV);
        } else if (tid == 1 && Svalid > 1) {
            myA = ctc_log2(row[0][myExt] + EPSV);
        }
        if (tid >= Svalid) myA = NEGV;
        cur[tid] = myA;
    }

    // One DP step: consume row[t & (NBUF-1)], cur -> nxt, then swap.
#define CTC_STEP(t_)                                                         \
    do {                                                                     \
        if (tid < CTC_S) {                                                   \
            float a  = myA;                                                  \
            float bv = (tid >= 1) ? cur[tid - 1] : NEGV;                     \
            float cv = (tid >= 2 && skipOk) ? cur[tid - 2] : NEGV;           \
            float m   = fmaxf(fmaxf(a, bv), cv);                             \
            float sum = ctc_exp2(a - m) + ctc_exp2(bv - m) + ctc_exp2(cv - m);\
            float e   = ctc_log2(row[(t_) & (NBUF - 1)][myExt] + EPSV);      \
            float nv  = m + ctc_log2(sum) + e;                               \
            if (tid >= Svalid) nv = NEGV;                                    \
            if ((t_) < il) myA = nv;   /* freeze past input_length */        \
            nxt[tid] = myA;                                                  \
        }                                                                    \
        float* tmp_ = cur; cur = nxt; nxt = tmp_;                            \
    } while (0)

    // Main loop: 7 loads in flight; only the oldest must have completed.
    for (int t = 1; t <= CTC_T - NBUF + 1; ++t) {
        ctc_wait_async_le6();             // row t done (7 outstanding max)
        __syncthreads();                  // ...and alpha from t-1 visible
        if (t < CTC_T - NBUF + 1) {       // prefetch row t+7 over row t-1's slot
            ctc_async_ldg_b128(
                ctc_lds_off(&row[(t + NBUF - 1) & (NBUF - 1)][tid * 4]),
                gcol + (size_t)(t + NBUF - 1) * CTC_V);
        }
        CTC_STEP(t);
    }

    // Drain tail: fewer than 7 loads outstanding -> wait them all.
    for (int t = CTC_T - NBUF + 2; t < CTC_T; ++t) {
        ctc_wait_async0();
        __syncthreads();
        CTC_STEP(t);
    }
#undef CTC_STEP

    __syncthreads();
    if (tid == 0) {
        float x = cur[2 * ll - 1];        // last label lane
        float y = cur[2 * ll];            // trailing blank lane
        float m = fmaxf(x, y);
        float loglik2 = m + ctc_log2(ctc_exp2(x - m) + ctc_exp2(y - m));
        loss_ws[b] = -loglik2 * LN2F;     // back to natural log
    }
}

// Deterministic mean over B per-utterance losses (no float atomics).
__global__ void ctc_mean_kernel(const float* __restrict__ loss_ws,
                                float* __restrict__ out)
{
    if (threadIdx.x == 0 && blockIdx.x == 0) {
        float s = 0.0f;
        for (int i = 0; i < CTC_B; ++i) s += loss_ws[i];
        out[0] = s / (float)CTC_B;
    }
}

extern "C" void kernel_launch(void* const* d_in, const int* in_sizes, int n_in,
                              void* d_out, int out_size, void* d_ws, size_t ws_size,
                              hipStream_t stream) {
    (void)in_sizes; (void)n_in; (void)out_size; (void)ws_size;

    const float* predictions  = (const float*)d_in[0]; // [B,T,V]
    const int*   input_lens   = (const int*)  d_in[1]; // [B,1]
    const int*   labels       = (const int*)  d_in[2]; // [B,L]
    const int*   label_lens   = (const int*)  d_in[3]; // [B,1]
    float*       out          = (float*)d_out;         // scalar mean loss
    float*       per_utt      = (float*)d_ws;          // B floats scratch

    ctc_alpha_kernel<<<dim3(CTC_B), dim3(256), 0, stream>>>(
        predictions, input_lens, labels, label_lens, per_utt);

    ctc_mean_kernel<<<dim3(1), dim3(32), 0, stream>>>(per_utt, out);
}